// LinkUtilAwareLoss_5549097747152
// MI455X (gfx1250) — compile-verified
//
#include <hip/hip_runtime.h>

typedef __attribute__((ext_vector_type(2))) float v2f;
typedef __attribute__((ext_vector_type(8))) float v8f;

#define TT 1020   // tunnels
#define DDN 255   // destinations (= K-chunks of 4 tunnels)
#define LLN 16    // links
#define SEGS 5    // K-split factor: 255 = 5 x 51 (equal segments, no tail branch)
#define CPS 51    // chunks per segment (compile-time constant trip count)
#define EPSF 1e-8f

// One K=4 chunk: A = tunnel_traffic fragment (16x4 f32), B = one-hot (4x16 f32).
// Lane layout (wave32): hi = lane>>4, ncol = lane&15.
//   A: lane -> row M = ncol, K pair = {2*hi, 2*hi+1}  (contiguous float2 in memory)
//   B: lane -> col N = ncol, K pair = {2*hi, 2*hi+1}  (one-hot vs tunnel_to_link)
__device__ __forceinline__ v8f chunk_mma(const float* __restrict__ pA,
                                         const float* __restrict__ pD,
                                         const int* __restrict__ pL,
                                         int k, int ncol, v8f c)
{
    float2 pr = *(const float2*)(pA + 4 * k);   // pred_ratios[row][4k+2*hi .. +1]
    float  d  = pD[k];                          // demands[row][k] (dst == chunk, K==4)
    int2   tl = *(const int2*)(pL + 4 * k);     // tunnel_to_link[4k+2*hi .. +1]
    v2f a; a.x = pr.x * d;  a.y = pr.y * d;
    v2f b; b.x = (tl.x == ncol) ? 1.0f : 0.0f;
           b.y = (tl.y == ncol) ? 1.0f : 0.0f;
    return __builtin_amdgcn_wmma_f32_16x16x4_f32(
        /*neg_a=*/false, a, /*neg_b=*/false, b,
        /*c_mod=*/(short)0, c, /*reuse_a=*/false, /*reuse_b=*/false);
}

// Block = 160 threads = 5 wave32 = 1 batch tile x 5 equal K-segments.
__global__ __launch_bounds__(160) void lua_tile_kernel(
    const float* __restrict__ pred,   // [B, 1020]
    const float* __restrict__ dem,    // [B, 255]
    const float* __restrict__ clu,    // [B, 16]
    const float* __restrict__ caps,   // [16]
    const int*   __restrict__ t2l,    // [1020]
    float* __restrict__ partials)     // [B/16] per-tile sums of row losses
{
    __shared__ float smC[SEGS][256];  // [seg][j*32+lane], 5 KB

    const int lane = threadIdx.x & 31;
    const int seg  = threadIdx.x >> 5;        // wave id == K-segment id
    const int tile = blockIdx.x;

    const int hi   = lane >> 4;
    const int ncol = lane & 15;
    const int mrow = tile * 16 + ncol;

    const float* pA = pred + (size_t)mrow * TT + 2 * hi;
    const float* pD = dem  + (size_t)mrow * DDN;
    const int*   pL = t2l + 2 * hi;

    const int kBeg = seg * CPS;               // wave-uniform, constant length

    // 25 two-chunk iterations (two accumulators break the WMMA D->C chain)
    // plus one unconditional final chunk: exactly 51 chunks, no tail branch.
    v8f c0 = {}; v8f c1 = {};
    int k = kBeg;
    #pragma unroll 2
    for (int i = 0; i < CPS / 2; ++i, k += 2) {
        c0 = chunk_mma(pA, pD, pL, k,     ncol, c0);
        c1 = chunk_mma(pA, pD, pL, k + 1, ncol, c1);
    }
    c0 = chunk_mma(pA, pD, pL, kBeg + CPS - 1, ncol, c0);

    // Stash this segment's partial C tile in LDS.
    #pragma unroll
    for (int j = 0; j < 8; ++j)
        smC[seg][j * 32 + lane] = c0[j] + c1[j];
    __syncthreads();

    if (seg != 0) return;   // one wave per tile finishes the (cheap) stats

    // Combine the 5 K-segment partials.
    float c[8];
    #pragma unroll
    for (int j = 0; j < 8; ++j) {
        float v = 0.0f;
        #pragma unroll
        for (int s = 0; s < SEGS; ++s) v += smC[s][j * 32 + lane];
        c[j] = v;
    }

    // node_link_util = traffic / (cap + eps); per-row stats across the 16 lanes
    // of each half-wave (C vgpr j: rows j / j+8, col = lane&15).
    const float cap = caps[ncol];
    const float uc  = 1.0f / (cap + EPSF);

    float acc = 0.0f;   // sum of per-row losses handled by this half-wave
    #pragma unroll
    for (int j = 0; j < 8; ++j) {
        const int rowj = tile * 16 + j + 8 * hi;
        const float u  = c[j] * uc;
        const float cl = clu[(size_t)rowj * LLN + ncol];
        float s = u, s2 = u * u, mx = u, cg = u * cl;
        #pragma unroll
        for (int off = 1; off <= 8; off <<= 1) {
            s  += __shfl_xor(s,  off, 32);
            s2 += __shfl_xor(s2, off, 32);
            mx  = fmaxf(mx, __shfl_xor(mx, off, 32));
            cg += __shfl_xor(cg, off, 32);
        }
        // unbiased variance over 16 links (ddof=1)
        const float var = (s2 - s * s * (1.0f / 16.0f)) * (1.0f / 15.0f);
        acc += 0.3f * var + 0.5f * cg + 0.2f * mx;   // duplicated across half lanes
    }
    acc += __shfl_xor(acc, 16, 32);   // combine the two 8-row halves
    if (lane == 0) partials[tile] = acc;
}

__global__ __launch_bounds__(256) void lua_reduce_kernel(
    const float* __restrict__ partials, float* __restrict__ out, int n, float scale)
{
    __shared__ float sm[256];
    float s = 0.0f;
    for (int i = threadIdx.x; i < n; i += 256) s += partials[i];
    sm[threadIdx.x] = s;
    __syncthreads();
    for (int w = 128; w > 0; w >>= 1) {
        if ((int)threadIdx.x < w) sm[threadIdx.x] += sm[threadIdx.x + w];
        __syncthreads();
    }
    if (threadIdx.x == 0) out[0] = sm[0] * scale;
}

extern "C" void kernel_launch(void* const* d_in, const int* in_sizes, int n_in,
                              void* d_out, int out_size, void* d_ws, size_t ws_size,
                              hipStream_t stream) {
    const float* pred = (const float*)d_in[0];   // [B,1020]
    const float* dem  = (const float*)d_in[1];   // [B,255]
    const float* clu  = (const float*)d_in[2];   // [B,16]
    const float* caps = (const float*)d_in[3];   // [16]
    const int*   t2l  = (const int*)d_in[4];     // [1020]
    float* out = (float*)d_out;

    const int T  = in_sizes[4];          // 1020
    const int Bn = in_sizes[0] / T;      // 16384
    const int nTiles = Bn / 16;          // 1024 (B is a multiple of 16)
    float* partials = (float*)d_ws;      // nTiles floats

    // 1 tile per block: 5 waves = 5 equal K-segments of 51 chunks.
    dim3 block(32 * SEGS);
    dim3 grid(nTiles);
    lua_tile_kernel<<<grid, block, 0, stream>>>(pred, dem, clu, caps, t2l, partials);
    lua_reduce_kernel<<<1, 256, 0, stream>>>(partials, out, nTiles, 1.0f / (float)Bn);
}